// ClusterAverage_41738492182555
// MI455X (gfx1250) — compile-verified
//
#include <hip/hip_runtime.h>
#include <math.h>
#include <stdint.h>

// ---------------- problem constants (from reference) ----------------
constexpr int   NLEN_   = 500000;
constexpr int   D_      = 128;
constexpr int   BATCH_  = 64;
constexpr int   CLIPS_  = 4;
constexpr int   BS_     = 256;          // BATCH*CLIPS
constexpr int   NNEG_   = 252;          // (BATCH-1)*CLIPS
constexpr int   NCOL_   = 253;          // 1 + NNEG
constexpr int   NN_     = 8;
constexpr float T_      = 0.07f;

// tiling for the big GEMM
constexpr int   MTILE_   = 16;                          // memory rows per WMMA tile
constexpr int   TPG_     = 4;                           // tiles per group
constexpr int   GROW_    = MTILE_ * TPG_;               // 64 memory rows per group
constexpr int   NGROUPS_ = (NLEN_ + GROW_ - 1) / GROW_; // 7813
constexpr int   GSTRIDE_ = 7840;                        // padded row stride for blockmax

constexpr float NEG_INF_ = -3.4e38f;

// workspace layout (in floats)
constexpr size_t WS_BMAX_ = 0;                                   // 256*GSTRIDE
constexpr size_t WS_YI_   = WS_BMAX_ + (size_t)BS_ * GSTRIDE_;   // 2048 ints
constexpr size_t WS_POSL_ = WS_YI_ + BS_ * NN_;                  // 256 floats
constexpr size_t WS_E_    = WS_POSL_ + BS_;                      // 256*253 floats
constexpr size_t WS_RSUM_ = WS_E_ + (size_t)BS_ * NCOL_;         // 256 floats

// output layout (in floats)
constexpr size_t OUT_E_     = 0;                 // outs [256,253]
constexpr size_t OUT_PROBS_ = (size_t)BS_ * NCOL_;
constexpr size_t OUT_MEM_   = OUT_PROBS_ + 1;    // new_memory [500000,128]

typedef float v2f_t __attribute__((ext_vector_type(2)));
typedef float v8f_t __attribute__((ext_vector_type(8)));

#define LDS_MSTRIDE 132   // 16 mem rows, padded to kill LDS bank conflicts

// ---- CDNA5 async global->LDS copy (ASYNCcnt path) -------------------
// global_load_async_to_lds_b128: per-lane copy of 16B from global memory
// straight into LDS (no VGPR round trip).  Low 32 bits of a flat LDS
// pointer are the LDS byte offset (ISA 10.2 aperture rules).
__device__ __forceinline__ void async_copy_b128(uint32_t lds_addr,
                                                const float* gaddr) {
  asm volatile("global_load_async_to_lds_b128 %0, %1, off"
               :
               : "v"(lds_addr), "v"(gaddr)
               : "memory");
}
__device__ __forceinline__ void wait_async_le1() {
  asm volatile("s_wait_asynccnt 0x1" ::: "memory");
}
__device__ __forceinline__ void wait_async_le0() {
  asm volatile("s_wait_asynccnt 0x0" ::: "memory");
}

// =====================================================================
// Kernel 1: WMMA fp32 GEMM  sim = x @ memory^T, reduced to per-row
// blockmax over 64-column groups.  512 threads = 16 waves; wave w owns
// x-rows [16w,16w+16); A-fragments cached in 64 VGPRs/lane; memory
// tiles double-buffered in LDS via async global->LDS copies.
// =====================================================================
__global__ __launch_bounds__(512) void sim_blockmax_kernel(
    const float* __restrict__ x, const float* __restrict__ mem,
    float* __restrict__ blockmax) {
  __shared__ float mt[2][MTILE_ * LDS_MSTRIDE];

  const int g    = blockIdx.x;
  const int tid  = threadIdx.x;
  const int wave = tid >> 5;
  const int lane = tid & 31;
  const int half = lane >> 4;   // 0..1
  const int l16  = lane & 15;   // 0..15

  // ---- A fragments: 16x4 f32 layout: lanes 0-15 hold K={k0,k0+1},
  // lanes 16-31 hold K={k0+2,k0+3}; M = lane%16.  32 k-steps of 4.
  v2f_t afrag[32];
  {
    const float* xrow = x + (size_t)(wave * 16 + l16) * D_;
    #pragma unroll
    for (int ks = 0; ks < 32; ++ks)
      afrag[ks] = *(const v2f_t*)(xrow + ks * 4 + half * 2);
  }

  float rmax[8];
  #pragma unroll
  for (int r = 0; r < 8; ++r) rmax[r] = NEG_INF_;

  const long base_row = (long)g * GROW_;
  const int  ldr = tid >> 5;        // LDS tile row this thread copies (== wave)
  const int  ldc = lane * 4;        // 4-float column chunk

  // prologue: async copy tile 0 into buffer 0.
  // NOTE: OOB rows clamp the *address* to row 0 (never predicate the copy
  // off) so every wave's ASYNCcnt advances identically; junk data is
  // masked out by the col<NLEN check below.
  {
    long gr = base_row + ldr;
    if (gr >= NLEN_) gr = 0;
    async_copy_b128((uint32_t)(uintptr_t)&mt[0][ldr * LDS_MSTRIDE + ldc],
                    mem + gr * D_ + ldc);
  }

  for (int t = 0; t < TPG_; ++t) {
    const int cur = t & 1;
    if (t + 1 < TPG_) {
      long gr = base_row + (long)(t + 1) * MTILE_ + ldr;
      if (gr >= NLEN_) gr = 0;
      async_copy_b128(
          (uint32_t)(uintptr_t)&mt[cur ^ 1][ldr * LDS_MSTRIDE + ldc],
          mem + gr * D_ + ldc);
      wait_async_le1();   // tile t's (older) copy complete for this wave
    } else {
      wait_async_le0();
    }
    __syncthreads();      // all waves' portions of tile t visible

    // ---- preload all 32 B fragments, then a back-to-back WMMA chain
    v2f_t bfrag[32];
    {
      const float* bbase = &mt[cur][l16 * LDS_MSTRIDE + half * 2];
      #pragma unroll
      for (int ks = 0; ks < 32; ++ks)
        bfrag[ks] = *(const v2f_t*)(bbase + ks * 4);
    }

    v8f_t acc = {0.f, 0.f, 0.f, 0.f, 0.f, 0.f, 0.f, 0.f};
    #pragma unroll
    for (int ks = 0; ks < 32; ++ks) {
      acc = __builtin_amdgcn_wmma_f32_16x16x4_f32(
          /*neg_a=*/false, afrag[ks], /*neg_b=*/false, bfrag[ks],
          /*c_mod=*/(short)0, acc, /*reuse_a=*/false, /*reuse_b=*/false);
    }

    // ---- fold into per-row running max (mask OOB columns)
    const long col = base_row + (long)t * MTILE_ + l16;
    #pragma unroll
    for (int r = 0; r < 8; ++r) {
      float v = (col < NLEN_) ? acc[r] : NEG_INF_;
      v = fmaxf(v, __shfl_xor(v, 1, 32));
      v = fmaxf(v, __shfl_xor(v, 2, 32));
      v = fmaxf(v, __shfl_xor(v, 4, 32));
      v = fmaxf(v, __shfl_xor(v, 8, 32));
      rmax[r] = fmaxf(rmax[r], v);
    }
    __syncthreads();      // compute on buf[cur] done before it is re-filled
  }

  if (l16 == 0) {
    #pragma unroll
    for (int r = 0; r < 8; ++r) {
      const int row = wave * 16 + r + 8 * half;  // D-matrix M mapping
      blockmax[(size_t)row * GSTRIDE_ + g] = rmax[r];
    }
  }
}

// ---------------- small top-8 helpers ----------------
__device__ __forceinline__ void insert8(float v, int id, float* lv, int* li) {
  if (v <= lv[7]) return;
  int p = 7;
  while (p > 0 && lv[p - 1] < v) { lv[p] = lv[p - 1]; li[p] = li[p - 1]; --p; }
  lv[p] = v; li[p] = id;
}

__device__ __forceinline__ void merge_lds(float* sv, int* si, int dst, int src) {
  float av[8], bv[8]; int ai[8], bi[8];
  #pragma unroll
  for (int k = 0; k < 8; ++k) {
    av[k] = sv[dst * 8 + k]; ai[k] = si[dst * 8 + k];
    bv[k] = sv[src * 8 + k]; bi[k] = si[src * 8 + k];
  }
  float rv[8]; int ri[8]; int ia = 0, ib = 0;
  #pragma unroll
  for (int k = 0; k < 8; ++k) {
    const bool ta = av[ia] >= bv[ib];
    rv[k] = ta ? av[ia] : bv[ib];
    ri[k] = ta ? ai[ia] : bi[ib];
    if (ta) ++ia; else ++ib;
  }
  #pragma unroll
  for (int k = 0; k < 8; ++k) { sv[dst * 8 + k] = rv[k]; si[dst * 8 + k] = ri[k]; }
}

// =====================================================================
// Kernel 2: exact per-row top-8 from blockmax (threshold + recompute)
// one workgroup (256 threads) per x-row.
// =====================================================================
__global__ __launch_bounds__(256) void topk_kernel(
    const float* __restrict__ x, const float* __restrict__ mem,
    const float* __restrict__ bmax, int* __restrict__ yi,
    float* __restrict__ posl) {
  const int row = blockIdx.x, tid = threadIdx.x;
  __shared__ float xb[D_];
  __shared__ float sv[256 * 8];
  __shared__ int   si[256 * 8];
  __shared__ int   cnt;
  __shared__ int   cand[1024];

  if (tid < D_) xb[tid] = x[(size_t)row * D_ + tid];
  const float* bm = bmax + (size_t)row * GSTRIDE_;

  // pass 1: top-8 of blockmaxes -> threshold t8 (provable lower bound of v8)
  float lv[8]; int li[8];
  #pragma unroll
  for (int r = 0; r < 8; ++r) { lv[r] = NEG_INF_; li[r] = -1; }
  for (int g = tid; g < NGROUPS_; g += 256) {
    const float v = bm[g];
    if (v > lv[7]) insert8(v, g, lv, li);
  }
  #pragma unroll
  for (int r = 0; r < 8; ++r) { sv[tid * 8 + r] = lv[r]; si[tid * 8 + r] = li[r]; }
  __syncthreads();
  for (int s = 128; s >= 1; s >>= 1) {
    if (tid < s) merge_lds(sv, si, tid, tid + s);
    __syncthreads();
  }
  const float t8 = sv[7];

  // pass 2: collect candidate groups
  if (tid == 0) cnt = 0;
  __syncthreads();
  for (int g = tid; g < NGROUPS_; g += 256) {
    if (bm[g] >= t8) {
      const int p = atomicAdd(&cnt, 1);
      if (p < 1024) cand[p] = g;
    }
  }
  __syncthreads();
  const int nc = min(cnt, 1024);

  // pass 3: recompute exact dots for candidate columns, exact top-8
  #pragma unroll
  for (int r = 0; r < 8; ++r) { lv[r] = NEG_INF_; li[r] = -1; }
  for (int c = tid; c < nc * GROW_; c += 256) {
    const int col = cand[c >> 6] * GROW_ + (c & 63);
    if (col < NLEN_) {
      const float* mr = mem + (size_t)col * D_;
      float dot = 0.f;
      #pragma unroll 4
      for (int d = 0; d < D_; ++d) dot += xb[d] * mr[d];
      if (dot > lv[7]) insert8(dot, col, lv, li);
    }
  }
  __syncthreads();
  #pragma unroll
  for (int r = 0; r < 8; ++r) { sv[tid * 8 + r] = lv[r]; si[tid * 8 + r] = li[r]; }
  __syncthreads();
  for (int s = 128; s >= 1; s >>= 1) {
    if (tid < s) merge_lds(sv, si, tid, tid + s);
    __syncthreads();
  }
  if (tid == 0) {
    float s = 0.f;
    #pragma unroll
    for (int k = 0; k < 8; ++k) { yi[row * NN_ + k] = si[k]; s += sv[k]; }
    posl[row] = s * (1.f / (float)NN_);   // pos_logit == mean of top-8 sims
  }
}

// =====================================================================
// Kernel 3a: e = exp(logits/T) (unnormalized) + row sums
// =====================================================================
__global__ __launch_bounds__(256) void logits_kernel(
    const float* __restrict__ x, const int* __restrict__ negidx,
    const float* __restrict__ posl, float* __restrict__ e,
    float* __restrict__ rsum) {
  const int b = blockIdx.x, tid = threadIdx.x;
  __shared__ float xb[D_];
  __shared__ float red[256];
  if (tid < D_) xb[tid] = x[(size_t)b * D_ + tid];
  __syncthreads();

  float ev = 0.f;
  if (tid == 0) {
    ev = expf(posl[b] / T_);
    e[(size_t)b * NCOL_] = ev;
  } else if (tid <= NNEG_) {
    const int j = negidx[(size_t)b * NNEG_ + (tid - 1)];
    const float* xj = x + (size_t)j * D_;
    float dot = 0.f;
    #pragma unroll 4
    for (int d = 0; d < D_; ++d) dot += xb[d] * xj[d];
    ev = expf(dot / T_);
    e[(size_t)b * NCOL_ + tid] = ev;
  }
  red[tid] = ev;
  __syncthreads();
  for (int s = 128; s > 0; s >>= 1) {
    if (tid < s) red[tid] += red[tid + s];
    __syncthreads();
  }
  if (tid == 0) rsum[b] = red[0];
}

// =====================================================================
// Kernel 3b: Z, normalized outs, probs (single workgroup)
// =====================================================================
__global__ __launch_bounds__(256) void finalize_kernel(
    const float* __restrict__ e, const float* __restrict__ rsum,
    float* __restrict__ out) {
  const int tid = threadIdx.x;
  __shared__ float red[256];
  red[tid] = rsum[tid];
  __syncthreads();
  for (int s = 128; s > 0; s >>= 1) {
    if (tid < s) red[tid] += red[tid + s];
    __syncthreads();
  }
  const float Z = (red[0] / (float)(BS_ * NCOL_)) * (float)NLEN_;
  const float invZ = 1.f / Z;
  for (int i = tid; i < BS_ * NCOL_; i += 256) out[OUT_E_ + i] = e[i] * invZ;

  const float p = e[(size_t)tid * NCOL_] / rsum[tid];  // Z cancels
  __syncthreads();
  red[tid] = p;
  __syncthreads();
  for (int s = 128; s > 0; s >>= 1) {
    if (tid < s) red[tid] += red[tid + s];
    __syncthreads();
  }
  if (tid == 0) out[OUT_PROBS_] = red[0] / (float)BS_;
}

// =====================================================================
// Kernel 4: EMA memory update of the 64 idxs rows (after D2D bank copy)
// =====================================================================
__global__ __launch_bounds__(128) void memupd_kernel(
    const float* __restrict__ mem, const int* __restrict__ idxs,
    const int* __restrict__ yi, float* __restrict__ newmem) {
  const int b = blockIdx.x, d = threadIdx.x;
  float s = 0.f;
  #pragma unroll
  for (int c = 0; c < CLIPS_; ++c)
    #pragma unroll
    for (int k = 0; k < NN_; ++k) {
      const int r = yi[(c * BATCH_ + b) * NN_ + k];
      s += mem[(size_t)r * D_ + d];
    }
  const float nd = s * (1.f / (float)(CLIPS_ * NN_));
  const int t = idxs[b];
  const float old = mem[(size_t)t * D_ + d];
  const float u = 0.5f * old + 0.5f * nd;

  float sq = u * u;
  sq += __shfl_xor(sq, 16, 32);
  sq += __shfl_xor(sq, 8, 32);
  sq += __shfl_xor(sq, 4, 32);
  sq += __shfl_xor(sq, 2, 32);
  sq += __shfl_xor(sq, 1, 32);
  __shared__ float wsum[4];
  if ((d & 31) == 0) wsum[d >> 5] = sq;
  __syncthreads();
  const float tot = wsum[0] + wsum[1] + wsum[2] + wsum[3];
  newmem[(size_t)t * D_ + d] = u * rsqrtf(tot);
}

// =====================================================================
extern "C" void kernel_launch(void* const* d_in, const int* in_sizes, int n_in,
                              void* d_out, int out_size, void* d_ws, size_t ws_size,
                              hipStream_t stream) {
  const float* x      = (const float*)d_in[0];
  const int*   idxs   = (const int*)d_in[1];
  const int*   negidx = (const int*)d_in[2];
  const float* mem    = (const float*)d_in[3];
  // d_in[4] = i (unused)

  float* out  = (float*)d_out;
  float* ws   = (float*)d_ws;
  float* bmax = ws + WS_BMAX_;
  int*   yi   = (int*)(ws + WS_YI_);
  float* posl = ws + WS_POSL_;
  float* e    = ws + WS_E_;
  float* rsum = ws + WS_RSUM_;
  float* newmem = out + OUT_MEM_;

  // bulk copy of the memory bank into new_memory (same stream, serialized)
  hipMemcpyAsync(newmem, mem, (size_t)NLEN_ * D_ * sizeof(float),
                 hipMemcpyDeviceToDevice, stream);

  sim_blockmax_kernel<<<NGROUPS_, 512, 0, stream>>>(x, mem, bmax);
  topk_kernel<<<BS_, 256, 0, stream>>>(x, mem, bmax, yi, posl);
  logits_kernel<<<BS_, 256, 0, stream>>>(x, negidx, posl, e, rsum);
  finalize_kernel<<<1, 256, 0, stream>>>(e, rsum, out);
  memupd_kernel<<<BATCH_, 128, 0, stream>>>(mem, idxs, yi, newmem);
}